// GCN_69226282877028
// MI455X (gfx1250) — compile-verified
//
#include <hip/hip_runtime.h>
#include <math.h>

typedef __attribute__((ext_vector_type(2))) float v2f;
typedef __attribute__((ext_vector_type(8))) float v8f;

#define LN_EPS 1e-5f

__device__ __forceinline__ void atomAddF(float* p, float v) {
  unsafeAtomicAdd(p, v);   // lowers to global_atomic_add_f32 on gfx1250
}

// ---------------------------------------------------------------------------
// Degree / normalization kernels
// ---------------------------------------------------------------------------
__global__ void init_deg(float* deg, int n) {
  int i = blockIdx.x * blockDim.x + threadIdx.x;
  if (i < n) deg[i] = 1.0f;   // self-loop contribution
}

__global__ void deg_accum(const int* __restrict__ dst, float* deg, int E) {
  int i = blockIdx.x * blockDim.x + threadIdx.x;
  if (i < E) atomAddF(&deg[dst[i]], 1.0f);
}

__global__ void finish_dinv(float* deg, int n) {
  int i = blockIdx.x * blockDim.x + threadIdx.x;
  if (i < n) {
    float d = deg[i];
    deg[i] = (d > 0.0f) ? rsqrtf(d) : 0.0f;
  }
}

// ---------------------------------------------------------------------------
// fp32 WMMA GEMM: H[nrows,FOUT] = X[nrows,128] @ W[128,FOUT]
// One wave computes one 16x16 C tile via 32x V_WMMA_F32_16X16X4_F32.
// ---------------------------------------------------------------------------
template <int FOUT>
__global__ __launch_bounds__(256) void gemm_wmma_f32(
    const float* __restrict__ X, const float* __restrict__ W,
    float* __restrict__ H, int nrows)
{
  constexpr int TPR = FOUT / 16;   // column tiles per row-tile
  constexpr int RPB = 8 / TPR;     // row tiles per 8-wave block
  const int w    = threadIdx.x >> 5;
  const int lane = threadIdx.x & 31;

  const int rowTile = blockIdx.x * RPB + (w / TPR);
  const int colTile = w % TPR;
  const int r0 = rowTile * 16;
  if (r0 >= nrows) return;
  const int c0 = colTile * 16;

  const int lh = lane & 15;          // row (A) / column (B,C) within tile
  const int kh = (lane >> 4) << 1;   // K sub-offset: 0 for lanes 0-15, 2 for 16-31

  const float* __restrict__ xrow = X + (size_t)(r0 + lh) * 128 + kh;
  const float* __restrict__ wcol = W + (size_t)kh * FOUT + c0 + lh;

  v8f acc = {};
#pragma unroll
  for (int kk = 0; kk < 128; kk += 4) {
    v2f a; a.x = xrow[0];   a.y = xrow[1];        // A: 16x4 tile, K=kh,kh+1
    v2f b; b.x = wcol[0];   b.y = wcol[FOUT];     // B: 4x16 tile, K=kh,kh+1
    acc = __builtin_amdgcn_wmma_f32_16x16x4_f32(
        false, a, false, b, (short)0, acc, false, false);
    xrow += 4;
    wcol += 4 * FOUT;
  }

  // C/D layout: VGPR v -> row r0+v (lanes 0-15) / r0+v+8 (lanes 16-31)
  const int rbase = r0 + ((lane >> 4) << 3);
  float* hp = H + (size_t)rbase * FOUT + c0 + lh;
#pragma unroll
  for (int v = 0; v < 8; ++v) hp[(size_t)v * FOUT] = acc[v];
}

// ---------------------------------------------------------------------------
// Edge scatter: AGG[dst] += H[src] * dinv[src]*dinv[dst]; one wave per edge.
// ---------------------------------------------------------------------------
template <int FOUT>
__global__ __launch_bounds__(256) void scatter_edges(
    const float* __restrict__ H, const int* __restrict__ srcA,
    const int* __restrict__ dstA, const float* __restrict__ dinv,
    float* __restrict__ AGG, int E, int n)
{
  constexpr int VEC = FOUT / 32;  // floats per lane
  const int gw   = (int)((blockIdx.x * blockDim.x + threadIdx.x) >> 5);
  const int lane = threadIdx.x & 31;
  const int total = E + n;
  if (gw >= total) return;

  int s, d;
  if (gw < E) { s = srcA[gw]; d = dstA[gw]; }
  else        { s = gw - E;   d = s; }           // self-loop

  const float nrm = dinv[s] * dinv[d];
  const float* __restrict__ hp = H + (size_t)s * FOUT + lane * VEC;
  float* ap = AGG + (size_t)d * FOUT + lane * VEC;

  float v[VEC];
#pragma unroll
  for (int j = 0; j < VEC; ++j) v[j] = hp[j] * nrm;   // coalesces to b128/b64
#pragma unroll
  for (int j = 0; j < VEC; ++j) atomAddF(ap + j, v[j]);
}

// ---------------------------------------------------------------------------
// Fused bias + LayerNorm + ReLU, one wave per node (128 features, 4/lane).
// ---------------------------------------------------------------------------
__global__ __launch_bounds__(256) void ln_relu_128(
    const float* __restrict__ AGG, const float* __restrict__ bias,
    const float* __restrict__ gamma, const float* __restrict__ beta,
    float* __restrict__ OUT, int n)
{
  const int node = (int)((blockIdx.x * blockDim.x + threadIdx.x) >> 5);
  const int lane = threadIdx.x & 31;
  if (node >= n) return;

  const float* row = AGG + (size_t)node * 128 + lane * 4;
  float v0 = row[0] + bias[lane * 4 + 0];
  float v1 = row[1] + bias[lane * 4 + 1];
  float v2 = row[2] + bias[lane * 4 + 2];
  float v3 = row[3] + bias[lane * 4 + 3];

  float s = v0 + v1 + v2 + v3;
#pragma unroll
  for (int m = 16; m >= 1; m >>= 1) s += __shfl_xor(s, m, 32);
  const float mu = s * (1.0f / 128.0f);

  float d0 = v0 - mu, d1 = v1 - mu, d2 = v2 - mu, d3 = v3 - mu;
  float q = d0 * d0 + d1 * d1 + d2 * d2 + d3 * d3;
#pragma unroll
  for (int m = 16; m >= 1; m >>= 1) q += __shfl_xor(q, m, 32);
  const float rstd = rsqrtf(q * (1.0f / 128.0f) + LN_EPS);

  float* out = OUT + (size_t)node * 128 + lane * 4;
  out[0] = fmaxf(0.0f, d0 * rstd * gamma[lane * 4 + 0] + beta[lane * 4 + 0]);
  out[1] = fmaxf(0.0f, d1 * rstd * gamma[lane * 4 + 1] + beta[lane * 4 + 1]);
  out[2] = fmaxf(0.0f, d2 * rstd * gamma[lane * 4 + 2] + beta[lane * 4 + 2]);
  out[3] = fmaxf(0.0f, d3 * rstd * gamma[lane * 4 + 3] + beta[lane * 4 + 3]);
}

// Final layer: out = agg + bias (no LN/ReLU), 64 features.
__global__ void bias_add_64(const float* __restrict__ AGG,
                            const float* __restrict__ bias,
                            float* __restrict__ OUT, int total)
{
  int i = blockIdx.x * blockDim.x + threadIdx.x;
  if (i < total) OUT[i] = AGG[i] + bias[i & 63];
}

// ---------------------------------------------------------------------------
extern "C" void kernel_launch(void* const* d_in, const int* in_sizes, int n_in,
                              void* d_out, int out_size, void* d_ws, size_t ws_size,
                              hipStream_t stream) {
  const float* x   = (const float*)d_in[0];
  const int*   ei  = (const int*)d_in[1];
  const float* W1  = (const float*)d_in[2];
  const float* b1  = (const float*)d_in[3];
  const float* g1  = (const float*)d_in[4];
  const float* be1 = (const float*)d_in[5];
  const float* W2  = (const float*)d_in[6];
  const float* b2  = (const float*)d_in[7];
  const float* g2  = (const float*)d_in[8];
  const float* be2 = (const float*)d_in[9];
  const float* W3  = (const float*)d_in[10];
  const float* b3  = (const float*)d_in[11];

  const int N = in_sizes[0] / 128;
  const int E = in_sizes[1] / 2;
  const int* src = ei;
  const int* dst = ei + E;

  float* ws   = (float*)d_ws;
  float* dinv = ws;                        // [N]
  float* h    = dinv + N;                  // [N,128]
  float* agg  = h + (size_t)N * 128;       // [N,128]
  float* hn   = agg + (size_t)N * 128;     // [N,128]

  // --- normalization coefficients -----------------------------------------
  init_deg   <<<(N + 255) / 256, 256, 0, stream>>>(dinv, N);
  deg_accum  <<<(E + 255) / 256, 256, 0, stream>>>(dst, dinv, E);
  finish_dinv<<<(N + 255) / 256, 256, 0, stream>>>(dinv, N);

  const int total   = E + N;
  const int sblocks = (total + 7) / 8;     // 8 waves (edges) per 256-thread block

  // --- layer 1 -------------------------------------------------------------
  gemm_wmma_f32<128><<<(N + 15) / 16, 256, 0, stream>>>(x, W1, h, N);
  hipMemsetAsync(agg, 0, (size_t)N * 128 * sizeof(float), stream);
  scatter_edges<128><<<sblocks, 256, 0, stream>>>(h, src, dst, dinv, agg, E, N);
  ln_relu_128<<<(N + 7) / 8, 256, 0, stream>>>(agg, b1, g1, be1, hn, N);

  // --- layer 2 -------------------------------------------------------------
  gemm_wmma_f32<128><<<(N + 15) / 16, 256, 0, stream>>>(hn, W2, h, N);
  hipMemsetAsync(agg, 0, (size_t)N * 128 * sizeof(float), stream);
  scatter_edges<128><<<sblocks, 256, 0, stream>>>(h, src, dst, dinv, agg, E, N);
  ln_relu_128<<<(N + 7) / 8, 256, 0, stream>>>(agg, b2, g2, be2, hn, N);

  // --- layer 3 -------------------------------------------------------------
  gemm_wmma_f32<64><<<(N + 31) / 32, 256, 0, stream>>>(hn, W3, h, N);
  hipMemsetAsync(agg, 0, (size_t)N * 64 * sizeof(float), stream);
  scatter_edges<64><<<sblocks, 256, 0, stream>>>(h, src, dst, dinv, agg, E, N);
  bias_add_64<<<(N * 64 + 255) / 256, 256, 0, stream>>>(agg, b3, (float*)d_out, N * 64);
}